// GCN_79894981640830
// MI455X (gfx1250) — compile-verified
//
#include <hip/hip_runtime.h>

#define HDIM 128
#define NCLS 40
#define BN_EPS 1e-5f

typedef __attribute__((ext_vector_type(2))) float v2f;
typedef __attribute__((ext_vector_type(8))) float v8f;

// ---------------- degree / normalization ----------------
__global__ void k_set_ones(float* deg, int n) {
  int i = blockIdx.x * blockDim.x + threadIdx.x;
  if (i < n) deg[i] = 1.0f;  // self-loop contributes 1
}

__global__ void k_deg_accum(const int* __restrict__ ei, float* deg, int E) {
  int e = blockIdx.x * blockDim.x + threadIdx.x;
  if (e < E) atomicAdd(&deg[ei[E + e]], 1.0f);  // dst row of edge_index
}

__global__ void k_rsqrt_ip(float* d, int n) {
  int i = blockIdx.x * blockDim.x + threadIdx.x;
  if (i < n) d[i] = rsqrtf(d[i]);  // deg >= 1 always (self loops)
}

// ---------------- BN statistics (column sums) ----------------
__global__ void k_stats(const float* __restrict__ x, float* __restrict__ stats, int n) {
  // 256 threads: 2 threads per column
  int col = threadIdx.x & (HDIM - 1);
  int half = threadIdx.x >> 7;
  float s = 0.f, s2 = 0.f;
  for (int r = blockIdx.x * 2 + half; r < n; r += gridDim.x * 2) {
    float v = x[(size_t)r * HDIM + col];
    s += v; s2 += v * v;
  }
  __shared__ float ls[HDIM], ls2[HDIM];
  if (half == 0) { ls[col] = s; ls2[col] = s2; }
  __syncthreads();
  if (half == 1) {
    atomicAdd(&stats[col], s + ls[col]);
    atomicAdd(&stats[HDIM + col], s2 + ls2[col]);
  }
}

// add bias + relu in place; optionally accumulate next layer's BN stats
__global__ void k_bias_relu_stats(float* __restrict__ x, const float* __restrict__ b,
                                  float* __restrict__ stats, int n, int do_stats) {
  int col = threadIdx.x & (HDIM - 1);
  int half = threadIdx.x >> 7;
  float bias = b[col];
  float s = 0.f, s2 = 0.f;
  for (int r = blockIdx.x * 2 + half; r < n; r += gridDim.x * 2) {
    size_t idx = (size_t)r * HDIM + col;
    float v = x[idx] + bias;
    v = v > 0.f ? v : 0.f;
    x[idx] = v;
    s += v; s2 += v * v;
  }
  if (!do_stats) return;
  __shared__ float ls[HDIM], ls2[HDIM];
  if (half == 0) { ls[col] = s; ls2[col] = s2; }
  __syncthreads();
  if (half == 1) {
    atomicAdd(&stats[col], s + ls[col]);
    atomicAdd(&stats[HDIM + col], s2 + ls2[col]);
  }
}

// ---------------- fold BN into weights: WfT[n][k] = a[k]*W[k][n]; cb[n] = sum_k c[k]*W[k][n]
__global__ void k_fold(const float* __restrict__ stats, const float* __restrict__ gamma,
                       const float* __restrict__ beta, const float* __restrict__ W,
                       float* __restrict__ WfT, float* __restrict__ cb, float inv_n) {
  __shared__ float as[HDIM], cs[HDIM];
  int t = threadIdx.x;  // 128 threads, one block
  float mean = stats[t] * inv_n;
  float var  = stats[HDIM + t] * inv_n - mean * mean;
  float a = gamma[t] * rsqrtf(var + BN_EPS);
  float c = beta[t] - a * mean;
  as[t] = a; cs[t] = c;
  __syncthreads();
  float acc = 0.f;
  for (int k = 0; k < HDIM; ++k) {
    float w = W[k * HDIM + t];
    WfT[t * HDIM + k] = as[k] * w;
    acc += cs[k] * w;
  }
  cb[t] = acc;
}

// ---------------- WMMA f32 GEMM: h[N,128] = x_bn[N,128] @ Wf + cb ----------------
#define XPAD 68  // 64 + 4 pad floats -> bank stride 4, conflict-free b64 reads

__global__ __launch_bounds__(128) void k_gemm(const float* __restrict__ x,
                                              const float* __restrict__ WfT,
                                              const float* __restrict__ cb,
                                              float* __restrict__ h, int n) {
  __shared__ float xs[64 * XPAD];     // 64 rows x 64 k (half K)
  __shared__ float wfs[HDIM * XPAD];  // 128 n x 64 k (half K)
  const int tid = threadIdx.x;        // 0..127 (4 waves)
  const int lane = tid & 31;
  const int wave = tid >> 5;
  const int rowBase = blockIdx.x * 64;
  const int m = lane & 15;            // A row / B col / C col
  const int hi = lane >> 4;           // lane half

  v8f acc[8];
#pragma unroll
  for (int c = 0; c < 8; ++c)
#pragma unroll
    for (int v = 0; v < 8; ++v) acc[c][v] = 0.f;

  for (int kh = 0; kh < 2; ++kh) {
    // stage x tile: 64 rows x 64 floats, coalesced float4
#pragma unroll
    for (int p = 0; p < 8; ++p) {
      int idx = p * 512 + tid * 4;
      int r = idx >> 6, cc = idx & 63;
      int row = rowBase + r;
      float4 v = make_float4(0.f, 0.f, 0.f, 0.f);
      if (row < n) v = *(const float4*)(x + (size_t)row * HDIM + kh * 64 + cc);
      *(float4*)(&xs[r * XPAD + cc]) = v;
    }
    // stage WfT tile: 128 n x 64 floats
#pragma unroll
    for (int p = 0; p < 16; ++p) {
      int idx = p * 512 + tid * 4;
      int nn = idx >> 6, cc = idx & 63;
      *(float4*)(&wfs[nn * XPAD + cc]) = *(const float4*)(WfT + nn * HDIM + kh * 64 + cc);
    }
    __syncthreads();

    const float* xrow = &xs[(wave * 16 + m) * XPAD];
#pragma unroll
    for (int k4 = 0; k4 < 16; ++k4) {
      int kb = k4 * 4 + hi * 2;
      v2f a; a.x = xrow[kb]; a.y = xrow[kb + 1];
#pragma unroll
      for (int c = 0; c < 8; ++c) {
        const float* wrow = &wfs[(c * 16 + m) * XPAD];
        v2f b; b.x = wrow[kb]; b.y = wrow[kb + 1];
        acc[c] = __builtin_amdgcn_wmma_f32_16x16x4_f32(
            false, a, false, b, (short)0, acc[c], false, false);
      }
    }
    __syncthreads();
  }

  // epilogue: C element (vgpr v, lane) -> row = v + 8*hi, col = lane&15
#pragma unroll
  for (int c = 0; c < 8; ++c) {
    int col = c * 16 + m;
    float cbv = cb[col];
#pragma unroll
    for (int v = 0; v < 8; ++v) {
      int row = rowBase + wave * 16 + v + 8 * hi;
      if (row < n) h[(size_t)row * HDIM + col] = acc[c][v] + cbv;
    }
  }
}

// ---------------- edge-parallel scatter-add (one wave per edge) ----------------
__global__ void k_edge_agg(const int* __restrict__ ei, const float* __restrict__ dinv,
                           const float* __restrict__ hbuf, float* __restrict__ out,
                           int E, int n) {
  int e = blockIdx.x * 8 + (threadIdx.x >> 5);
  if (e >= E + n) return;
  int lane = threadIdx.x & 31;
  int src, dst;
  if (e < E) { src = ei[e]; dst = ei[E + e]; }
  else       { src = dst = e - E; }          // self loop
  float nrm = dinv[src] * dinv[dst];
  const float4 hv = *(const float4*)(hbuf + (size_t)src * HDIM + lane * 4);
  float* o = out + (size_t)dst * HDIM + lane * 4;
  atomicAdd(o + 0, hv.x * nrm);
  atomicAdd(o + 1, hv.y * nrm);
  atomicAdd(o + 2, hv.z * nrm);
  atomicAdd(o + 3, hv.w * nrm);
}

// ---------------- final projection: out[N,40] = x @ Wo + bo ----------------
__global__ __launch_bounds__(256) void k_out_gemm(const float* __restrict__ x,
                                                  const float* __restrict__ Wo,
                                                  const float* __restrict__ bo,
                                                  float* __restrict__ out, int n) {
  __shared__ float ws[HDIM * NCLS];
  __shared__ float bs[NCLS];
  int tid = threadIdx.x;
  for (int i = tid; i < HDIM * NCLS; i += 256) ws[i] = Wo[i];
  if (tid < NCLS) bs[tid] = bo[tid];
  __syncthreads();
  int r = blockIdx.x * 128 + (tid >> 1);
  int c0 = (tid & 1) * 20;
  if (r >= n) return;
  float acc[20];
#pragma unroll
  for (int c = 0; c < 20; ++c) acc[c] = bs[c0 + c];
  const float* xr = x + (size_t)r * HDIM;
  for (int k = 0; k < HDIM; ++k) {
    float xv = xr[k];
    const float* wr = &ws[k * NCLS + c0];
#pragma unroll
    for (int c = 0; c < 20; ++c) acc[c] += xv * wr[c];
  }
  float* o = out + (size_t)r * NCLS + c0;
#pragma unroll
  for (int c = 0; c < 20; ++c) o[c] = acc[c];
}

extern "C" void kernel_launch(void* const* d_in, const int* in_sizes, int n_in,
                              void* d_out, int out_size, void* d_ws, size_t ws_size,
                              hipStream_t stream) {
  const float* x_in  = (const float*)d_in[0];
  const int*   ei    = (const int*)d_in[1];
  const float* gamma = (const float*)d_in[2];
  const float* beta  = (const float*)d_in[3];
  const float* W     = (const float*)d_in[4];
  const float* b     = (const float*)d_in[5];
  const float* Wo    = (const float*)d_in[6];
  const float* bo    = (const float*)d_in[7];
  float* out = (float*)d_out;

  const int N = in_sizes[0] / HDIM;
  const int E = in_sizes[1] / 2;

  // workspace carve-up (float units)
  float* dinv  = (float*)d_ws;                        // N
  float* stats = dinv + ((N + 255) / 256) * 256;      // 2*H
  float* WfT   = stats + 2 * HDIM;                    // H*H
  float* cb    = WfT + HDIM * HDIM;                   // H
  float* hbuf  = cb + HDIM + 64;                      // N*H
  float* xbuf  = hbuf + (size_t)N * HDIM;             // N*H

  // degree -> dinv
  k_set_ones<<<(N + 255) / 256, 256, 0, stream>>>(dinv, N);
  k_deg_accum<<<(E + 255) / 256, 256, 0, stream>>>(ei, dinv, E);
  k_rsqrt_ip<<<(N + 255) / 256, 256, 0, stream>>>(dinv, N);

  // layer-0 BN stats on raw input
  hipMemsetAsync(stats, 0, 2 * HDIM * sizeof(float), stream);
  k_stats<<<512, 256, 0, stream>>>(x_in, stats, N);

  const int EN = E + N;
  for (int l = 0; l < 3; ++l) {
    k_fold<<<1, 128, 0, stream>>>(stats, gamma + l * HDIM, beta + l * HDIM,
                                  W + (size_t)l * HDIM * HDIM, WfT, cb, 1.0f / (float)N);
    const float* xl = (l == 0) ? x_in : xbuf;
    k_gemm<<<(N + 63) / 64, 128, 0, stream>>>(xl, WfT, cb, hbuf, N);
    // zero the segment-sum accumulator (bias folded into the relu pass)
    hipMemsetAsync(xbuf, 0, (size_t)N * HDIM * sizeof(float), stream);
    k_edge_agg<<<(EN + 7) / 8, 256, 0, stream>>>(ei, dinv, hbuf, xbuf, E, N);
    if (l < 2) {
      hipMemsetAsync(stats, 0, 2 * HDIM * sizeof(float), stream);
      k_bias_relu_stats<<<512, 256, 0, stream>>>(xbuf, b + l * HDIM, stats, N, 1);
    } else {
      k_bias_relu_stats<<<512, 256, 0, stream>>>(xbuf, b + l * HDIM, stats, N, 0);
    }
  }

  k_out_gemm<<<(N + 127) / 128, 256, 0, stream>>>(xbuf, Wo, bo, out, N);
}